// URSCT_22969485099066
// MI455X (gfx1250) — compile-verified
//
#include <hip/hip_runtime.h>
#include <hip/hip_bf16.h>

typedef __attribute__((ext_vector_type(2))) float v2f;
typedef __attribute__((ext_vector_type(8))) float v8f;

#define B_WIN   2048
#define NTOK    64
#define DIM     192
#define HEADS   6
#define HD      32
#define BN      (B_WIN * NTOK)      // 131072 tokens
#define C3      (3 * DIM)           // 576
#define SCALE_Q 0.17677669529663687f // 32^-0.5

// ---------------------------------------------------------------------------
// one f32 WMMA step: D(16x16) += A(16x4) * B(4x16)
// lane layout (ISA 7.12.2, 32-bit A): lane L holds M=L&15; VGPR0/1 = K = 2*(L>>4), +1
// B mirrored: lane L holds N=L&15; VGPR0/1 = K = 2*(L>>4), +1 (BT row-major, contiguous)
// ---------------------------------------------------------------------------
__device__ __forceinline__ v8f wmma_f32(v2f a, v2f b, v8f c) {
    return __builtin_amdgcn_wmma_f32_16x16x4_f32(false, a, false, b, (short)0, c,
                                                 false, false);
}

// ---------------------------------------------------------------------------
// K0: bias_full[h][i][j] = rpb[rel_index[i*64+j]*HEADS + h]
// ---------------------------------------------------------------------------
__global__ __launch_bounds__(256) void bias_gather_kernel(
    const float* __restrict__ rpb, const int* __restrict__ rel,
    float* __restrict__ biasf) {
    int f = blockIdx.x * 256 + threadIdx.x;
    if (f < HEADS * NTOK * NTOK) {
        int h = f / (NTOK * NTOK);
        int r = f % (NTOK * NTOK);
        biasf[f] = rpb[rel[r] * HEADS + h];
    }
}

// ---------------------------------------------------------------------------
// 8x8 real cosine transform G (== Re(fft2) of real input; /64 == Re(ifft2))
//   fwd:  out[(b*64+pix)*192 + c]          (token-major, feeds GEMM as A)
//   inv:  out[b*12288 + c*64 + pixout]     (flat == (b, head, t, d) remap, identity)
// ---------------------------------------------------------------------------
__global__ __launch_bounds__(256) void dft_fwd_kernel(
    const float* __restrict__ x, float* __restrict__ xg) {
    __shared__ float xb[NTOK * DIM]; // 48 KB: [pix][c]
    const long base = (long)blockIdx.x * (NTOK * DIM);
    for (int f = threadIdx.x; f < NTOK * DIM; f += 256) xb[f] = x[base + f];
    __syncthreads();
    const float ct[8] = {1.f, 0.70710678118654752f, 0.f, -0.70710678118654752f,
                         -1.f, -0.70710678118654752f, 0.f, 0.70710678118654752f};
    for (int f = threadIdx.x; f < NTOK * DIM; f += 256) {
        int c = f % DIM, pix = f / DIM;
        int hh = pix >> 3, ww = pix & 7;
        float acc = 0.f;
        for (int p = 0; p < 8; ++p) {
            int hp = (hh * p) & 7;
            const float* xr = xb + p * 8 * DIM + c;
            for (int q = 0; q < 8; ++q)
                acc += ct[(hp + ww * q) & 7] * xr[q * DIM];
        }
        xg[base + f] = acc;
    }
}

__global__ __launch_bounds__(256) void dft_inv_kernel(
    const float* __restrict__ z2, float* __restrict__ ifr) {
    __shared__ float xb[NTOK * DIM]; // [pix][c] token-major input
    const long base = (long)blockIdx.x * (NTOK * DIM);
    for (int f = threadIdx.x; f < NTOK * DIM; f += 256) xb[f] = z2[base + f];
    __syncthreads();
    const float ct[8] = {1.f, 0.70710678118654752f, 0.f, -0.70710678118654752f,
                         -1.f, -0.70710678118654752f, 0.f, 0.70710678118654752f};
    for (int f = threadIdx.x; f < NTOK * DIM; f += 256) {
        int c = f >> 6, pixo = f & 63;
        int hh = pixo >> 3, ww = pixo & 7;
        float acc = 0.f;
        for (int p = 0; p < 8; ++p) {
            int hp = (hh * p) & 7;
            const float* xr = xb + p * 8 * DIM + c;
            for (int q = 0; q < 8; ++q)
                acc += ct[(hp + ww * q) & 7] * xr[q * DIM];
        }
        ifr[base + f] = acc * (1.0f / 64.0f);
    }
}

// ---------------------------------------------------------------------------
// fp32 WMMA GEMM, register-blocked 64x32 wave tile (4 m-tiles x 2 n-tiles):
//   D[m][n] = sum_k A[m*K+k] * BT[n*K+k]   (+bias, +gelu), out[m*ldc + n]
// Each A fragment reused 2x, each B fragment 4x -> 8 WMMAs per 6 b64 loads.
// Jobs ordered n-fastest within an m-group so the 8 waves of a block share
// the same 64-row A panel through WGP$.
// mode bit0 = exact GELU epilogue, bit1 = bias indexed by n (else by m).
// Requires M % 64 == 0, N % 32 == 0, K % 4 == 0.
// ---------------------------------------------------------------------------
__global__ __launch_bounds__(256) void gemm_f32_wmma_kernel(
    const float* __restrict__ A, const float* __restrict__ BT,
    float* __restrict__ out, const float* __restrict__ bias,
    int M, int N, int K, int ldc, int mode) {
    const int wave = threadIdx.x >> 5;
    const int lane = threadIdx.x & 31;
    const int half = lane >> 4;
    const int l    = lane & 15;
    const int koff = half * 2;
    const int nGroups = N >> 5;
    const long job = (long)blockIdx.x * 8 + wave;
    if (job >= (long)(M >> 6) * nGroups) return;
    const int mg = (int)(job / nGroups);
    const int ng = (int)(job % nGroups);

    const float* a0 = A  + ((long)mg * 64 + l) * K;
    const float* b0 = BT + ((long)ng * 32 + l) * K;

    // get the tail of the K panels in flight early (global_prefetch_b8)
    #pragma unroll
    for (int i = 0; i < 4; ++i) __builtin_prefetch(a0 + i * 16 * K + K / 2, 0, 1);
    #pragma unroll
    for (int j = 0; j < 2; ++j) __builtin_prefetch(b0 + j * 16 * K + K / 2, 0, 1);

    v8f acc[4][2] = {};
    for (int k0 = 0; k0 < K; k0 += 4) {
        v2f a[4], b[2];
        #pragma unroll
        for (int i = 0; i < 4; ++i) a[i] = *(const v2f*)(a0 + i * 16 * K + k0 + koff);
        #pragma unroll
        for (int j = 0; j < 2; ++j) b[j] = *(const v2f*)(b0 + j * 16 * K + k0 + koff);
        #pragma unroll
        for (int i = 0; i < 4; ++i)
            #pragma unroll
            for (int j = 0; j < 2; ++j)
                acc[i][j] = wmma_f32(a[i], b[j], acc[i][j]);
    }

    #pragma unroll
    for (int i = 0; i < 4; ++i) {
        #pragma unroll
        for (int j = 0; j < 2; ++j) {
            const int n = ng * 32 + j * 16 + l;
            #pragma unroll
            for (int v = 0; v < 8; ++v) {
                int m = mg * 64 + i * 16 + v + 8 * half;
                float val = acc[i][j][v];
                if (bias) val += (mode & 2) ? bias[n] : bias[m];
                if (mode & 1)
                    val = 0.5f * val * (1.0f + erff(val * 0.70710678118654752f));
                out[(long)m * ldc + n] = val;
            }
        }
    }
}

// ---------------------------------------------------------------------------
// Fused per-(window,head) kernel:
//   depthwise 3x3 conv over the (2048 x 64) grid from channel-major t0,
//   + ifft add (+SCALE on q) -> q/k/v in LDS -> S=qK^T (WMMA) + bias ->
//   softmax -> O=PV (WMMA) -> token-major staging for the proj GEMM.
// ---------------------------------------------------------------------------
__global__ __launch_bounds__(256) void attn_fused_kernel(
    const float* __restrict__ t0, const float* __restrict__ ifr,
    const float* __restrict__ dww, const float* __restrict__ dwb,
    const float* __restrict__ biasf, float* __restrict__ aout) {
    const int b = blockIdx.x / HEADS;
    const int h = blockIdx.x % HEADS;
    __shared__ float qs[NTOK * HD];
    __shared__ float ks[NTOK * HD];
    __shared__ float vs[NTOK * HD];
    __shared__ float ss[NTOK * NTOK];

    // ---- phase 1: depthwise conv + ifft add (6144 outputs, t fastest) ----
    for (int f = threadIdx.x; f < 3 * NTOK * HD; f += 256) {
        int t  = f & 63;
        int sd = f >> 6;            // 0..95
        int s  = sd >> 5;           // qkv set
        int d  = sd & 31;
        int ch = s * DIM + h * HD + d;
        const float* w = dww + ch * 9;
        float acc = dwb[ch];
        #pragma unroll
        for (int db = -1; db <= 1; ++db) {
            int bb = b + db;
            if (bb < 0 || bb >= B_WIN) continue;
            const float* row = t0 + (long)ch * BN + (long)bb * NTOK;
            #pragma unroll
            for (int dn = -1; dn <= 1; ++dn) {
                int tt = t + dn;
                if (tt < 0 || tt >= NTOK) continue;
                acc += row[tt] * w[(db + 1) * 3 + (dn + 1)];
            }
        }
        float iv = ifr[(long)b * (HEADS * NTOK * HD) + h * (NTOK * HD) + t * HD + d];
        float val = acc + iv;
        if (s == 0) val *= SCALE_Q;
        float* dst = (s == 0) ? qs : (s == 1) ? ks : vs;
        dst[t * HD + d] = val;
    }
    __syncthreads();

    const int wave = threadIdx.x >> 5;
    const int lane = threadIdx.x & 31;
    const int half = lane >> 4;
    const int l    = lane & 15;
    const int koff = half * 2;

    // ---- phase 2: S = q * k^T (64x64, K=32) + relpos bias; 2 tiles/wave ----
    #pragma unroll
    for (int rep = 0; rep < 2; ++rep) {
        int tile = wave * 2 + rep;
        int tm = tile >> 2, tn = tile & 3;
        const float* qrow = qs + (tm * 16 + l) * HD;
        const float* krow = ks + (tn * 16 + l) * HD;
        v8f acc = {};
        #pragma unroll
        for (int k0 = 0; k0 < HD; k0 += 4) {
            v2f av = *(const v2f*)(qrow + k0 + koff);
            v2f bv = *(const v2f*)(krow + k0 + koff);
            acc = wmma_f32(av, bv, acc);
        }
        int ng = tn * 16 + l;
        #pragma unroll
        for (int v = 0; v < 8; ++v) {
            int m = tm * 16 + v + 8 * half;
            ss[m * NTOK + ng] = acc[v] + biasf[h * (NTOK * NTOK) + m * NTOK + ng];
        }
    }
    __syncthreads();

    // ---- phase 3: row softmax ----
    if (threadIdx.x < NTOK) {
        float* row = ss + threadIdx.x * NTOK;
        float mx = row[0];
        for (int j = 1; j < NTOK; ++j) mx = fmaxf(mx, row[j]);
        float sum = 0.f;
        for (int j = 0; j < NTOK; ++j) { float e = __expf(row[j] - mx); row[j] = e; sum += e; }
        float inv = 1.0f / sum;
        for (int j = 0; j < NTOK; ++j) row[j] *= inv;
    }
    __syncthreads();

    // ---- phase 4: O = P * V (64x32, K=64); 1 tile/wave ----
    {
        int tm = wave >> 1, tn = wave & 1;
        const float* prow = ss + (tm * 16 + l) * NTOK;
        v8f acc = {};
        #pragma unroll
        for (int k0 = 0; k0 < NTOK; k0 += 4) {
            v2f av = *(const v2f*)(prow + k0 + koff);
            v2f bv;
            bv.x = vs[(k0 + koff)     * HD + tn * 16 + l];
            bv.y = vs[(k0 + koff + 1) * HD + tn * 16 + l];
            acc = wmma_f32(av, bv, acc);
        }
        int ng = tn * 16 + l;
        #pragma unroll
        for (int v = 0; v < 8; ++v) {
            int m = tm * 16 + v + 8 * half; // token
            aout[((long)b * NTOK + m) * DIM + h * HD + ng] = acc[v];
        }
    }
}

// ---------------------------------------------------------------------------
extern "C" void kernel_launch(void* const* d_in, const int* in_sizes, int n_in,
                              void* d_out, int out_size, void* d_ws, size_t ws_size,
                              hipStream_t stream) {
    (void)in_sizes; (void)n_in; (void)out_size; (void)ws_size;
    const float* x      = (const float*)d_in[0];
    const float* qkv_w  = (const float*)d_in[1];
    const float* qkv_b  = (const float*)d_in[2];
    const float* dw_w   = (const float*)d_in[3];
    const float* dw_b   = (const float*)d_in[4];
    const float* q1_w   = (const float*)d_in[5];
    const float* q2_w   = (const float*)d_in[6];
    const float* rpb    = (const float*)d_in[7];
    const float* proj_w = (const float*)d_in[8];
    const float* proj_b = (const float*)d_in[9];
    const int*   relidx = (const int*)d_in[10];
    float* out = (float*)d_out;

    // workspace layout (floats); R1/R2 regions reused across stages
    float* biasf = (float*)d_ws;                        // 24576
    float* R1    = biasf + HEADS * NTOK * NTOK;         // 25,165,824 (xg -> z2 -> attnout)
    float* R2    = R1 + (long)BN * DIM;                 // 25,165,824 (z1 -> ifft)
    float* t0    = R2 + (long)BN * DIM;                 // 75,497,472 (576 x 131072)

    // wave-job counts for the 64x32-tile GEMM (8 waves / block)
    const int jobs_tok = (BN >> 6) * (DIM >> 5);  // M=131072, N=192 -> 12288
    const int jobs_qkv = (C3 >> 6) * (BN >> 5);   // M=576,    N=131072 -> 36864

    // 1) relative-position bias table
    bias_gather_kernel<<<(HEADS * NTOK * NTOK + 255) / 256, 256, 0, stream>>>(
        rpb, relidx, biasf);

    // 2) forward cosine transform (Re fft2)  x -> xg (R1, token-major)
    dft_fwd_kernel<<<B_WIN, 256, 0, stream>>>(x, R1);

    // 3) z1 = GELU(q1_w @ xg)  -> R2  (M=tokens, N=192, BT=q1_w)
    gemm_f32_wmma_kernel<<<jobs_tok / 8, 256, 0, stream>>>(
        R1, q1_w, R2, nullptr, BN, DIM, DIM, DIM, /*gelu*/1);

    // 4) z2 = q2_w @ z1 -> R1
    gemm_f32_wmma_kernel<<<jobs_tok / 8, 256, 0, stream>>>(
        R2, q2_w, R1, nullptr, BN, DIM, DIM, DIM, 0);

    // 5) inverse transform /64 -> ifft (R2, flat (b, head, t, d))
    dft_inv_kernel<<<B_WIN, 256, 0, stream>>>(R1, R2);

    // 6) QKV 1x1: t0[ch][token] = qkv_w @ x^T + qkv_b   (M=576, N=131072)
    gemm_f32_wmma_kernel<<<jobs_qkv / 8, 256, 0, stream>>>(
        qkv_w, x, t0, qkv_b, C3, BN, DIM, BN, 0);

    // 7) fused dwconv + ifft-add + attention -> attnout (R1, token-major)
    attn_fused_kernel<<<B_WIN * HEADS, 256, 0, stream>>>(
        t0, R2, dw_w, dw_b, biasf, R1);

    // 8) projection: out = attnout @ proj_w^T + proj_b (bias per-n)
    gemm_f32_wmma_kernel<<<jobs_tok / 8, 256, 0, stream>>>(
        R1, proj_w, out, proj_b, BN, DIM, DIM, DIM, /*bias per-n*/2);
}